// Aggregate_Att_Mean_89945205113500
// MI455X (gfx1250) — compile-verified
//
#include <hip/hip_runtime.h>
#include <hip/hip_bf16.h>

// ---------------- problem constants ----------------
#define NB    2048          // B
#define NS1   10            // S1
#define NS2   25            // S2
#define ND    128           // N_DIM = L10 = L11
#define SROW  (1 + NS1 + NS1*NS2)   // 261
#define NPAIR (NB * NS1)    // 20480

typedef __attribute__((ext_vector_type(2))) float v2f;
typedef __attribute__((ext_vector_type(8))) float v8f;

__device__ __forceinline__ float wave_reduce_sum(float v) {
  #pragma unroll
  for (int off = 16; off > 0; off >>= 1) v += __shfl_xor(v, off, 32);
  return v;
}
__device__ __forceinline__ float leaky(float x) { return x >= 0.0f ? x : 0.01f * x; }
__device__ __forceinline__ float dot4(float4 a, float4 b) {
  return a.x*b.x + a.y*b.y + a.z*b.z + a.w*b.w;
}

// ---------------- K0: precompute a1,a2,b1,b2 and Wc = [Wl@V1_h1 | Wr@V1_h1] -----
__global__ void precompute_kernel(const float* __restrict__ V1_h1_att,
                                  const float* __restrict__ w1_h1,
                                  const float* __restrict__ V1_h0_att,
                                  const float* __restrict__ w1_h0,
                                  const float* __restrict__ V1_h1,
                                  const float* __restrict__ W_hops,
                                  float* __restrict__ consts,   // 512 floats
                                  float* __restrict__ Wc) {     // 128*256 floats
  int tid = blockIdx.x * blockDim.x + threadIdx.x;
  if (tid < 512) {
    int which = tid >> 7;           // 0:a1 1:a2 2:b1 3:b2
    int n = tid & 127;
    const float* V = (which < 2) ? V1_h1_att : V1_h0_att;
    const float* w = (which < 2) ? w1_h1 : w1_h0;
    int wo = (which & 1) * 128;
    float s = 0.0f;
    #pragma unroll 4
    for (int l = 0; l < 128; ++l) s += V[l*128 + n] * w[wo + l];
    consts[tid] = s;
  } else if (tid < 512 + 128*256) {
    int t = tid - 512;
    int d = t >> 8;                 // output row (0..127)
    int k = t & 255;                // output col (0..255)
    const float* wh = W_hops + d*256 + ((k & 128) ? 128 : 0);
    int kk = k & 127;
    float s = 0.0f;
    #pragma unroll 4
    for (int l = 0; l < 128; ++l) s += wh[l] * V1_h1[l*128 + kk];
    Wc[d*256 + k] = s;
  }
}

// ---------------- K1: level-2 attention, one wave per (b,s) pair ----------------
__global__ void __launch_bounds__(256)
att2_kernel(const float* __restrict__ x, const int* __restrict__ samples,
            const float* __restrict__ consts,
            float* __restrict__ A2,       // (NPAIR, 256): [fea_t1 | wsum2]
            float* __restrict__ d2,       // (NPAIR)
            float* __restrict__ betaOut)  // d_out + 128*2048, (NPAIR, 26)
{
  int wid  = threadIdx.x >> 5;
  int lane = threadIdx.x & 31;
  int pair = blockIdx.x * 8 + wid;
  if (pair >= NPAIR) return;
  int b = pair / NS1, s = pair % NS1;
  const int* srow = samples + (size_t)b * SROW;

  float4 a1c = *(const float4*)(consts +       lane*4);
  float4 a2c = *(const float4*)(consts + 128 + lane*4);

  int i1 = srow[1 + s];
  float4 ft = *(const float4*)(x + (size_t)i1 * ND + lane*4);
  float c1 = wave_reduce_sum(dot4(ft, a1c));   // fea_t1 . a1
  float dv = wave_reduce_sum(dot4(ft, a2c));   // fea_t1 . a2  (for level-1 att)
  if (lane == 0) d2[pair] = dv;
  *(float4*)(A2 + (size_t)pair * 256 + lane*4) = ft;

  float4 rows[NS2];
  float  lg[NS2];
  const int* nidx = srow + 1 + NS1 + s * NS2;
  #pragma unroll
  for (int t = 0; t < NS2; ++t) {
    int in = nidx[t];
    float4 r = *(const float4*)(x + (size_t)in * ND + lane*4);
    rows[t] = r;
    lg[t] = c1 + wave_reduce_sum(dot4(r, a2c));
  }
  float mx = -3.0e38f;
  #pragma unroll
  for (int t = 0; t < NS2; ++t) { lg[t] = leaky(lg[t]); mx = fmaxf(mx, lg[t]); }
  float se = 0.0f;
  #pragma unroll
  for (int t = 0; t < NS2; ++t) { lg[t] = __expf(lg[t] - mx); se += lg[t]; }
  float inv = 1.0f / se;
  float4 wsum = make_float4(0.f, 0.f, 0.f, 0.f);
  #pragma unroll
  for (int t = 0; t < NS2; ++t) {
    float bta = lg[t] * inv;
    wsum.x += bta * rows[t].x; wsum.y += bta * rows[t].y;
    wsum.z += bta * rows[t].z; wsum.w += bta * rows[t].w;
  }
  *(float4*)(A2 + (size_t)pair * 256 + 128 + lane*4) = wsum;
  if (lane < NS2) betaOut[(size_t)pair * 26 + 1 + lane] = lg[lane] * inv;
}

// ---------------- K1b: level-1 attention, one wave per b ----------------
__global__ void __launch_bounds__(256)
att1_kernel(const float* __restrict__ x, const int* __restrict__ samples,
            const float* __restrict__ consts, const float* __restrict__ d2,
            float* __restrict__ A0)  // (NB, 256): [fea_t0 | fwsum0]
{
  int wid  = threadIdx.x >> 5;
  int lane = threadIdx.x & 31;
  int b = blockIdx.x * 8 + wid;
  if (b >= NB) return;
  const int* srow = samples + (size_t)b * SROW;

  float4 a1c = *(const float4*)(consts + lane*4);
  int i0 = srow[0];
  float4 f0 = *(const float4*)(x + (size_t)i0 * ND + lane*4);
  float c0 = wave_reduce_sum(dot4(f0, a1c));
  *(float4*)(A0 + (size_t)b * 256 + lane*4) = f0;

  float lg[NS1];
  float mx = -3.0e38f;
  #pragma unroll
  for (int s = 0; s < NS1; ++s) {
    lg[s] = leaky(c0 + d2[b * NS1 + s]);
    mx = fmaxf(mx, lg[s]);
  }
  float se = 0.0f;
  #pragma unroll
  for (int s = 0; s < NS1; ++s) { lg[s] = __expf(lg[s] - mx); se += lg[s]; }
  float inv = 1.0f / se;
  float4 wsum = make_float4(0.f, 0.f, 0.f, 0.f);
  #pragma unroll
  for (int s = 0; s < NS1; ++s) {
    int i1 = srow[1 + s];
    float4 r = *(const float4*)(x + (size_t)i1 * ND + lane*4);
    float bta = lg[s] * inv;
    wsum.x += bta * r.x; wsum.y += bta * r.y; wsum.z += bta * r.z; wsum.w += bta * r.w;
  }
  *(float4*)(A0 + (size_t)b * 256 + 128 + lane*4) = wsum;
}

// ---------------- WMMA FP32 GEMM: C[m,n] = act( sum_k A[m,k] * W[n,k] ) --------
// A: (M, K) row-major.  W: (128, K) row-major.  One wave -> one 16x16 C tile.
// v_wmma_f32_16x16x4_f32 fragment layouts per CDNA5 ISA 7.12.2.
template<int K, bool SIGMOID, bool TRANS>
__global__ void __launch_bounds__(128)
gemm_wmma_kernel(const float* __restrict__ A, const float* __restrict__ W,
                 float* __restrict__ C, int ldt)
{
  int lane = threadIdx.x & 31;
  int m0 = blockIdx.x * 16;
  int n0 = (blockIdx.y * blockDim.y + threadIdx.y) * 16;
  int half = lane >> 4;       // 0: K pair {0,1}; 1: K pair {2,3}
  int l15  = lane & 15;
  const float* Arow = A + (size_t)(m0 + l15) * K + half * 2;
  const float* Wrow = W + (size_t)(n0 + l15) * K + half * 2;

  v8f acc = {};
  #pragma unroll 8
  for (int k0 = 0; k0 < K; k0 += 4) {
    v2f af = *(const v2f*)(Arow + k0);
    v2f bf = *(const v2f*)(Wrow + k0);
    acc = __builtin_amdgcn_wmma_f32_16x16x4_f32(
        /*neg_a=*/false, af, /*neg_b=*/false, bf,
        /*c_mod=*/(short)0, acc, /*reuse_a=*/false, /*reuse_b=*/false);
  }

  int mbase = m0 + half * 8;
  int n = n0 + l15;
  #pragma unroll
  for (int r = 0; r < 8; ++r) {
    float v = acc[r];
    if (SIGMOID) v = 1.0f / (1.0f + __expf(-v));
    int m = mbase + r;
    if (TRANS) C[(size_t)n * ldt + m] = v;       // out is (128, M): fea_out.T
    else       C[(size_t)m * 128 + n] = v;
  }
}

// ---------------- K3: hop attention, one wave per b ----------------
__global__ void __launch_bounds__(256)
hopatt_kernel(const float* __restrict__ hop0, const float* __restrict__ hop1,
              const float* __restrict__ consts,
              float* __restrict__ hwsum,    // (NB, 128)
              float* __restrict__ betaOut)  // (NPAIR, 26), column 0
{
  int wid  = threadIdx.x >> 5;
  int lane = threadIdx.x & 31;
  int b = blockIdx.x * 8 + wid;
  if (b >= NB) return;

  float4 b1c = *(const float4*)(consts + 256 + lane*4);
  float4 b2c = *(const float4*)(consts + 384 + lane*4);

  float4 h0 = *(const float4*)(hop0 + (size_t)b * ND + lane*4);
  float c = wave_reduce_sum(dot4(h0, b1c));

  float4 rows[NS1];
  float  lg[NS1];
  #pragma unroll
  for (int s = 0; s < NS1; ++s) {
    float4 r = *(const float4*)(hop1 + ((size_t)b * NS1 + s) * ND + lane*4);
    rows[s] = r;
    lg[s] = c + wave_reduce_sum(dot4(r, b2c));
  }
  float mx = -3.0e38f;
  #pragma unroll
  for (int s = 0; s < NS1; ++s) { lg[s] = leaky(lg[s]); mx = fmaxf(mx, lg[s]); }
  float se = 0.0f;
  #pragma unroll
  for (int s = 0; s < NS1; ++s) { lg[s] = __expf(lg[s] - mx); se += lg[s]; }
  float inv = 1.0f / se;
  float4 wsum = make_float4(0.f, 0.f, 0.f, 0.f);
  #pragma unroll
  for (int s = 0; s < NS1; ++s) {
    float bta = lg[s] * inv;
    wsum.x += bta * rows[s].x; wsum.y += bta * rows[s].y;
    wsum.z += bta * rows[s].z; wsum.w += bta * rows[s].w;
  }
  hwsum[(size_t)b * ND + lane*4 + 0] = wsum.x;
  hwsum[(size_t)b * ND + lane*4 + 1] = wsum.y;
  hwsum[(size_t)b * ND + lane*4 + 2] = wsum.z;
  hwsum[(size_t)b * ND + lane*4 + 3] = wsum.w;
  if (lane < NS1) betaOut[((size_t)b * NS1 + lane) * 26] = lg[lane] * inv;
}

// ---------------- host-side orchestration ----------------
extern "C" void kernel_launch(void* const* d_in, const int* in_sizes, int n_in,
                              void* d_out, int out_size, void* d_ws, size_t ws_size,
                              hipStream_t stream) {
  const float* x        = (const float*)d_in[0];
  const int*   samples  = (const int*)  d_in[1];
  const float* V1h1att  = (const float*)d_in[2];
  const float* w1h1     = (const float*)d_in[3];
  const float* V1h0att  = (const float*)d_in[4];
  const float* w1h0     = (const float*)d_in[5];
  const float* V1h1     = (const float*)d_in[6];
  const float* V1h0     = (const float*)d_in[7];
  const float* Whops    = (const float*)d_in[8];

  float* out     = (float*)d_out;
  float* betaOut = out + (size_t)ND * NB;   // beta_step region, (NPAIR, 26)

  // workspace layout (floats)
  float* ws     = (float*)d_ws;
  float* consts = ws;                        // 512
  float* Wc     = consts + 512;              // 128*256 = 32768
  float* d2     = Wc + 128*256;              // NPAIR = 20480
  float* A2     = d2 + NPAIR;                // NPAIR*256 = 5242880
  float* A0     = A2 + (size_t)NPAIR * 256;  // NB*256   = 524288
  float* hop1   = A0 + (size_t)NB * 256;     // NPAIR*128 = 2621440
  float* hop0   = hop1 + (size_t)NPAIR * ND; // NB*128   = 262144
  float* hwsum  = hop0 + (size_t)NB * ND;    // NB*128   = 262144
  // total ~ 8.97M floats ~= 35.9 MB

  // K0: tiny precompute (512 + 32768 outputs)
  precompute_kernel<<<(512 + 128*256 + 255) / 256, 256, 0, stream>>>(
      V1h1att, w1h1, V1h0att, w1h0, V1h1, Whops, consts, Wc);

  // K1: level-2 attention (20480 waves)
  att2_kernel<<<NPAIR / 8, 256, 0, stream>>>(x, samples, consts, A2, d2, betaOut);

  // K1b: level-1 attention (2048 waves)
  att1_kernel<<<NB / 8, 256, 0, stream>>>(x, samples, consts, d2, A0);

  // K2: hop projections with sigmoid (WMMA fp32)
  gemm_wmma_kernel<256, true, false>
      <<<dim3(NPAIR / 16, 2), dim3(32, 4), 0, stream>>>(A2, Wc, hop1, 0);
  gemm_wmma_kernel<256, true, false>
      <<<dim3(NB / 16, 2), dim3(32, 4), 0, stream>>>(A0, Wc, hop0, 0);

  // K3: hop attention (2048 waves)
  hopatt_kernel<<<NB / 8, 256, 0, stream>>>(hop0, hop1, consts, hwsum, betaOut);

  // K4: fea_out.T = (V1_h0 @ hwsum.T)  -> transposed store into d_out[0 : 128*2048]
  gemm_wmma_kernel<128, false, true>
      <<<dim3(NB / 16, 2), dim3(32, 4), 0, stream>>>(hwsum, V1h0, out, NB);
}